// hd_rp_channel_encoder_11115375362812
// MI455X (gfx1250) — compile-verified
//
#include <hip/hip_runtime.h>

typedef float v2f __attribute__((ext_vector_type(2)));
typedef float v8f __attribute__((ext_vector_type(8)));

#define B_DIM 128
#define H_DIM 128
#define W_DIM 512
#define D_DIM 10000
#define NTILES (D_DIM / 16)   // 625
#define MTILES (B_DIM / 16)   // 8

// ---------------------------------------------------------------------------
// Kernel 1: xs[b][w] = sum_h x[b][h][w]   (memory-bound, float4 coalesced)
// ---------------------------------------------------------------------------
__global__ void __launch_bounds__(256) hd_reduce_h_kernel(
    const float* __restrict__ x, float* __restrict__ xs) {
  int t = blockIdx.x * blockDim.x + threadIdx.x;   // over B*W/4 = 16384
  int b  = t / (W_DIM / 4);
  int w4 = (t % (W_DIM / 4)) * 4;
  if (b >= B_DIM) return;

  const float4* xp =
      reinterpret_cast<const float4*>(x + (size_t)b * H_DIM * W_DIM + w4);
  float4 acc = make_float4(0.f, 0.f, 0.f, 0.f);
#pragma unroll 4
  for (int h = 0; h < H_DIM; ++h) {
    float4 v = xp[(size_t)h * (W_DIM / 4)];
    acc.x += v.x; acc.y += v.y; acc.z += v.z; acc.w += v.w;
  }
  *reinterpret_cast<float4*>(xs + (size_t)b * W_DIM + w4) = acc;
}

// ---------------------------------------------------------------------------
// Kernel 2: out = sign(xs @ W) via V_WMMA_F32_16X16X4_F32 (exact f32 path).
// One wave per 16-column tile of D; all 128 rows of C kept in registers
// (8 x v8f accumulators) so each B fragment is reused by 8 WMMAs.
// ---------------------------------------------------------------------------
__global__ void __launch_bounds__(128) hd_wmma_gemm_sign_kernel(
    const float* __restrict__ xs, const float* __restrict__ hdw,
    float* __restrict__ out) {
  const int lane  = threadIdx.x & 31;
  const int wave  = threadIdx.x >> 5;
  const int ntile = blockIdx.x * 4 + wave;
  if (ntile >= NTILES) return;                 // wave-uniform; EXEC stays full

  const int n0   = ntile * 16;
  const int half = lane >> 4;                  // 0: lanes 0-15, 1: lanes 16-31
  const int lmod = lane & 15;

  v8f c[MTILES];
#pragma unroll
  for (int g = 0; g < MTILES; ++g) c[g] = (v8f){0.f,0.f,0.f,0.f,0.f,0.f,0.f,0.f};

  // A frag (16x4 f32): lane half selects K-pair {0,1} vs {2,3}, lmod = M row.
  // B frag (4x16 f32): lmod = N col, half selects K-pair.
  for (int k0 = 0; k0 < W_DIM; k0 += 4) {
    const int kb = k0 + half * 2;

    v2f bfrag;
    bfrag.x = hdw[(size_t)kb       * D_DIM + n0 + lmod];
    bfrag.y = hdw[(size_t)(kb + 1) * D_DIM + n0 + lmod];

#pragma unroll
    for (int g = 0; g < MTILES; ++g) {
      const int m = g * 16 + lmod;
      v2f afrag = *reinterpret_cast<const v2f*>(xs + (size_t)m * W_DIM + kb);
      // 8 args: (neg_a, A, neg_b, B, c_mod, C, reuse_a, reuse_b)
      c[g] = __builtin_amdgcn_wmma_f32_16x16x4_f32(
          false, afrag, false, bfrag, (short)0, c[g], false, false);
    }
  }

  // C/D layout: VGPR v holds row M=v (lanes 0-15) or M=v+8 (lanes 16-31).
#pragma unroll
  for (int g = 0; g < MTILES; ++g) {
#pragma unroll
    for (int v = 0; v < 8; ++v) {
      const int m = g * 16 + v + half * 8;
      const float val = c[g][v];
      const float s = (val > 0.f) ? 1.f : ((val < 0.f) ? -1.f : 0.f);
      out[(size_t)m * D_DIM + n0 + lmod] = s;
    }
  }
}

// ---------------------------------------------------------------------------
extern "C" void kernel_launch(void* const* d_in, const int* in_sizes, int n_in,
                              void* d_out, int out_size, void* d_ws,
                              size_t ws_size, hipStream_t stream) {
  const float* x   = (const float*)d_in[0];   // (B,H,W) f32
  const float* hdw = (const float*)d_in[1];   // (W,D)   f32
  float* out = (float*)d_out;                 // (B,D)   f32
  float* xs  = (float*)d_ws;                  // (B,W)   f32 scratch, 256 KB

  // Stage 1: reduce over H.
  const int n1 = B_DIM * W_DIM / 4;           // 16384 threads
  hd_reduce_h_kernel<<<(n1 + 255) / 256, 256, 0, stream>>>(x, xs);

  // Stage 2: WMMA GEMM + sign. 625 column tiles, 4 waves (128 thr) per block.
  const int nblocks = (NTILES + 3) / 4;       // 157
  hd_wmma_gemm_sign_kernel<<<nblocks, 128, 0, stream>>>(xs, hdw, out);
}